// CRFDecoder_43473658970586
// MI455X (gfx1250) — compile-verified
//
#include <hip/hip_runtime.h>
#include <hip/hip_bf16.h>

typedef __attribute__((ext_vector_type(2))) float v2f;
typedef __attribute__((ext_vector_type(8))) float v8f;

#define NEGV  (-10000.0f)
#define Bsz   64
#define Ssz   512
#define Dsz   768
#define NLs   9
#define Lsz   11      // NL + 2
#define NPAD  16      // N padded to one WMMA tile
#define CSTART (Lsz - 2)
#define CEND   (Lsz - 1)

// ---------------------------------------------------------------------------
// Kernel 1: logits[m][0..15] = inputs[m,:] . W^T + bias  (cols >= 9 -> NEG)
// M = B*S = 32768 rows, K = 768. One wave per 16-row tile, fp32 WMMA 16x16x4.
// ---------------------------------------------------------------------------
__global__ __launch_bounds__(256) void crf_logits_wmma(
    const float* __restrict__ inp, const float* __restrict__ W,
    const float* __restrict__ bias, float* __restrict__ logits)
{
    // Pair-interleaved W^T staged in LDS: lwt[(kk*16 + n)*2 + j] = W[n][2*kk+j]
    __shared__ float lwt[(Dsz / 2) * NPAD * 2];   // 48 KB
    const int tid = threadIdx.x;
    for (int idx = tid; idx < (Dsz / 2) * NPAD * 2; idx += 256) {
        int j  = idx & 1;
        int n  = (idx >> 1) & 15;
        int kk = idx >> 5;
        int k  = 2 * kk + j;
        lwt[idx] = (n < NLs) ? W[n * Dsz + k] : 0.0f;
    }
    __syncthreads();

    const int lane  = tid & 31;
    const int wave  = tid >> 5;
    const int half  = lane >> 4;     // which half of the wave
    const int n     = lane & 15;     // N column (B/C layout) and M row (A layout)
    const int baseM = (blockIdx.x * 8 + wave) * 16;

    const float* arow = inp + (size_t)(baseM + n) * Dsz + 2 * half;

    v8f acc = {};
    #pragma unroll 4
    for (int k0 = 0; k0 < Dsz; k0 += 4) {
        if ((k0 & 63) == 0)
            __builtin_prefetch(arow + k0 + 256, 0, 1);   // global_prefetch_b8
        v2f a  = *(const v2f*)(arow + k0);                              // A: K = k0+2*half+{0,1}
        v2f bv = *(const v2f*)(&lwt[((k0 / 2 + half) * NPAD + n) * 2]); // B: same K pair, col n
        acc = __builtin_amdgcn_wmma_f32_16x16x4_f32(
            /*neg_a=*/false, a, /*neg_b=*/false, bv,
            /*c_mod=*/(short)0, acc, /*reuse_a=*/false, /*reuse_b=*/false);
    }

    const float bn = (n < NLs) ? bias[n] : 0.0f;
    #pragma unroll
    for (int v = 0; v < 8; ++v) {
        int m = baseM + half * 8 + v;                 // C layout: VGPR v -> row v (+8 for hi half)
        float val = (n < NLs) ? (acc[v] + bn) : NEGV; // classes 9..15 are NEG (start/end/unused)
        logits[(size_t)m * NPAD + n] = val;
    }
}

// ---------------------------------------------------------------------------
// Kernel 2: per-batch CRF forward scan + unary + binary + norm.
// One wave32 per batch. Lane i owns class i; full alpha vector replicated in
// registers per lane; alpha rebroadcast via shuffles each step.
// ---------------------------------------------------------------------------
__global__ __launch_bounds__(32) void crf_scan(
    const float* __restrict__ logits, const int* __restrict__ pmask,
    const int* __restrict__ labels, const float* __restrict__ trans,
    float* __restrict__ loglik)
{
    const int b    = blockIdx.x;
    const int lane = threadIdx.x;

    // length = sum(predict_mask[b,:])
    int len = 0;
    for (int t = lane; t < Ssz; t += 32) len += pmask[b * Ssz + t];
    for (int o = 16; o > 0; o >>= 1) len += __shfl_xor(len, o, 32);

    // unary + binary terms, parallel over t, butterfly-reduced
    float unary = 0.0f, binary = 0.0f;
    for (int t = lane; t < Ssz; t += 32) {
        if (t < len)
            unary += logits[((size_t)(b * Ssz + t)) * NPAD + labels[b * Ssz + t]];
    }
    for (int t = lane; t <= len; t += 32) {
        int cur  = (t == len) ? CEND   : labels[b * Ssz + t];
        int prev = (t == 0)   ? CSTART : labels[b * Ssz + t - 1];
        binary += trans[cur * Lsz + prev];
    }
    for (int o = 16; o > 0; o >>= 1) {
        unary  += __shfl_xor(unary, o, 32);
        binary += __shfl_xor(binary, o, 32);
    }

    // forward scan: alpha_{t+1}[i] = logit_t[i] + logsumexp_j(alpha_t[j] + trans[i][j])
    const int i  = lane;
    const int ii = (i < Lsz) ? i : 0;
    float tr[Lsz];
    #pragma unroll
    for (int j = 0; j < Lsz; ++j) tr[j] = trans[ii * Lsz + j];
    float al[Lsz];
    #pragma unroll
    for (int j = 0; j < Lsz; ++j) al[j] = (j == CSTART) ? 0.0f : NEGV;

    #pragma unroll 1
    for (int t = 0; t < Ssz; ++t) {
        if (t >= len) break;   // alpha frozen past the sequence end (len is wave-uniform)
        const int ci = (i < NPAD) ? i : 0;
        float lg = logits[((size_t)(b * Ssz + t)) * NPAD + ci];
        if (i >= Lsz) lg = NEGV;

        float m = 4.0f * NEGV;
        float v[Lsz];
        #pragma unroll
        for (int j = 0; j < Lsz; ++j) { v[j] = al[j] + tr[j]; m = fmaxf(m, v[j]); }
        float s = 0.0f;
        #pragma unroll
        for (int j = 0; j < Lsz; ++j) s += __expf(v[j] - m);
        float anew = lg + m + __logf(s);
        #pragma unroll
        for (int j = 0; j < Lsz; ++j) al[j] = __shfl(anew, j, 32);
    }

    // norm = logsumexp_i(alpha[i] + trans[END][i])
    float m2 = 4.0f * NEGV;
    float w[Lsz];
    #pragma unroll
    for (int j = 0; j < Lsz; ++j) { w[j] = al[j] + trans[CEND * Lsz + j]; m2 = fmaxf(m2, w[j]); }
    float s2 = 0.0f;
    #pragma unroll
    for (int j = 0; j < Lsz; ++j) s2 += __expf(w[j] - m2);
    float norm = m2 + __logf(s2);

    if (lane == 0) loglik[b] = unary + binary - norm;
}

// ---------------------------------------------------------------------------
// Kernel 3: loss = -mean(loglik)
// ---------------------------------------------------------------------------
__global__ __launch_bounds__(64) void crf_reduce(const float* __restrict__ loglik,
                                                 float* __restrict__ out)
{
    __shared__ float sh[Bsz];
    sh[threadIdx.x] = loglik[threadIdx.x];
    __syncthreads();
    if (threadIdx.x == 0) {
        float s = 0.0f;
        for (int k = 0; k < Bsz; ++k) s += sh[k];
        out[0] = -s / (float)Bsz;
    }
}

extern "C" void kernel_launch(void* const* d_in, const int* in_sizes, int n_in,
                              void* d_out, int out_size, void* d_ws, size_t ws_size,
                              hipStream_t stream)
{
    const float* inp  = (const float*)d_in[0];   // (B,S,D) f32
    const int*   pm   = (const int*)  d_in[1];   // (B,S) i32
    const int*   lb   = (const int*)  d_in[2];   // (B,S) i32
    const float* W    = (const float*)d_in[3];   // (NL,D) f32
    const float* bias = (const float*)d_in[4];   // (NL,) f32
    const float* tr   = (const float*)d_in[5];   // (L,L) f32

    float* logits = (float*)d_ws;                                   // 32768*16 f32 = 2 MB
    float* loglik = (float*)((char*)d_ws +
                             (size_t)Bsz * Ssz * NPAD * sizeof(float));

    crf_logits_wmma<<<(Bsz * Ssz) / 128, 256, 0, stream>>>(inp, W, bias, logits);
    crf_scan<<<Bsz, 32, 0, stream>>>(logits, pm, lb, tr, loglik);
    crf_reduce<<<1, 64, 0, stream>>>(loglik, (float*)d_out);
}